// SGCNet_90675349553257
// MI455X (gfx1250) — compile-verified
//
#include <hip/hip_runtime.h>
#include <math.h>

typedef float v2f __attribute__((ext_vector_type(2)));
typedef float v8f __attribute__((ext_vector_type(8)));

#define IN_CH 128
#define OUT_CH 40
#define NPAD 48

// ---------------- degree / norm ----------------
__global__ void deg_init_kernel(float* deg, int n) {
    int i = blockIdx.x * blockDim.x + threadIdx.x;
    if (i < n) deg[i] = 1.0f;  // self loop contributes 1
}

__global__ void deg_accum_kernel(const long long* __restrict__ col, float* deg, long long e) {
    long long i = (long long)blockIdx.x * blockDim.x + threadIdx.x;
    if (i < e) atomicAdd(&deg[(int)col[i]], 1.0f);
}

__global__ void rsqrt_kernel(float* deg, int n) {
    int i = blockIdx.x * blockDim.x + threadIdx.x;
    if (i < n) deg[i] = rsqrtf(deg[i]);  // deg >= 1 always
}

// ---------------- propagation ----------------
// out[i][:] = dis[i]^2 * in[i][:]   (self-loop term; also zero-initializes out)
__global__ void self_loop_kernel(const float* __restrict__ in, const float* __restrict__ dis,
                                 float* __restrict__ out, int n) {
    long long i = (long long)blockIdx.x * blockDim.x + threadIdx.x;  // one float4 each
    long long total = (long long)n * (IN_CH / 4);
    if (i >= total) return;
    int node = (int)(i >> 5);  // 32 float4 per node
    float s = dis[node];
    s *= s;
    float4 v = ((const float4*)in)[i];
    float4 o;
    o.x = s * v.x; o.y = s * v.y; o.z = s * v.z; o.w = s * v.w;
    ((float4*)out)[i] = o;
}

// one wave32 per edge: lane l handles channels [4l, 4l+4)
__global__ void spmm_edge_kernel(const long long* __restrict__ row, const long long* __restrict__ col,
                                 const float* __restrict__ dis, const float* __restrict__ hin,
                                 float* __restrict__ hout, long long e) {
    int lane = threadIdx.x & 31;
    int wid  = __builtin_amdgcn_readfirstlane(threadIdx.x >> 5);  // wave-uniform
    long long eid = (long long)blockIdx.x * 8 + wid;
    if (eid >= e) return;
    int r = (int)row[eid];
    int c = (int)col[eid];
    float nrm = dis[r] * dis[c];
    float4 v = ((const float4*)(hin + (size_t)r * IN_CH))[lane];  // coalesced 512B/wave
    float* dst = hout + (size_t)c * IN_CH + (size_t)(lane * 4);
    atomicAdd(dst + 0, nrm * v.x);
    atomicAdd(dst + 1, nrm * v.y);
    atomicAdd(dst + 2, nrm * v.z);
    atomicAdd(dst + 3, nrm * v.w);
}

// ---------------- fused GEMM (f32 WMMA) + bias + log_softmax ----------------
// grid: one wave per 16-row tile, 4 waves (128 threads) per block.
__global__ void __launch_bounds__(128) gemm_lsm_kernel(const float* __restrict__ h,
                                                       const float* __restrict__ W,
                                                       const float* __restrict__ bias,
                                                       float* __restrict__ out, int n) {
    __shared__ float Wt[NPAD * IN_CH];    // W transposed [n][k], cols 40..47 zero-padded
    __shared__ float L[4 * 16 * NPAD];    // per-wave 16x48 logits tile

    int tid = threadIdx.x;
    for (int i = tid; i < NPAD * IN_CH; i += 128) {
        int nn = i / IN_CH;
        int k  = i % IN_CH;
        Wt[i] = (nn < OUT_CH) ? W[(size_t)k * OUT_CH + nn] : 0.0f;
    }
    __syncthreads();

    int wave = tid >> 5;
    int lane = tid & 31;
    int tile = blockIdx.x * 4 + wave;
    if (tile * 16 >= n) return;  // whole-wave guard (EXEC stays all-ones for WMMA)

    int mrow = tile * 16 + (lane & 15);
    if (mrow >= n) mrow = n - 1;               // safety clamp (n is multiple of 16 here)
    int kb = (lane >> 4) * 2;                  // A/B: lanes 16-31 hold K+2,K+3
    const float* hrow = h + (size_t)mrow * IN_CH + kb;
    const float* bp0 = &Wt[((lane & 15) +  0) * IN_CH + kb];
    const float* bp1 = &Wt[((lane & 15) + 16) * IN_CH + kb];
    const float* bp2 = &Wt[((lane & 15) + 32) * IN_CH + kb];

    v8f c0 = {}, c1 = {}, c2 = {};
#pragma unroll
    for (int k = 0; k < IN_CH; k += 4) {
        v2f a  = *(const v2f*)(hrow + k);
        v2f f0 = *(const v2f*)(bp0 + k);
        v2f f1 = *(const v2f*)(bp1 + k);
        v2f f2 = *(const v2f*)(bp2 + k);
        c0 = __builtin_amdgcn_wmma_f32_16x16x4_f32(false, a, false, f0, (short)0, c0, false, false);
        c1 = __builtin_amdgcn_wmma_f32_16x16x4_f32(false, a, false, f1, (short)0, c1, false, false);
        c2 = __builtin_amdgcn_wmma_f32_16x16x4_f32(false, a, false, f2, (short)0, c2, false, false);
    }

    // Spill D tiles to wave-private LDS (same-wave LDS ops are in-order).
    float* lw = &L[wave * 16 * NPAD];
    int mbase = (lane >> 4) * 8;   // VGPR r: M = r (lanes 0-15) or r+8 (lanes 16-31)
    int nn = lane & 15;
#pragma unroll
    for (int r2 = 0; r2 < 8; ++r2) {
        lw[(mbase + r2) * NPAD + nn +  0] = c0[r2];
        lw[(mbase + r2) * NPAD + nn + 16] = c1[r2];
        lw[(mbase + r2) * NPAD + nn + 32] = c2[r2];
    }

    // lanes 0..15: one output row each -> bias + log_softmax over 40 logits
    if (lane < 16) {
        long long orow = (long long)tile * 16 + lane;
        if (orow < n) {
            const float* lp = &lw[lane * NPAD];
            float mx = -3.0e38f;
            for (int q = 0; q < OUT_CH; ++q) mx = fmaxf(mx, lp[q] + bias[q]);
            float s = 0.0f;
            for (int q = 0; q < OUT_CH; ++q) s += __expf(lp[q] + bias[q] - mx);
            float lse = mx + __logf(s);
            float* op = out + (size_t)orow * OUT_CH;
            for (int q = 0; q < OUT_CH; ++q) op[q] = lp[q] + bias[q] - lse;
        }
    }
}

extern "C" void kernel_launch(void* const* d_in, const int* in_sizes, int n_in,
                              void* d_out, int out_size, void* d_ws, size_t ws_size,
                              hipStream_t stream) {
    const float*     x    = (const float*)d_in[0];
    const long long* ei   = (const long long*)d_in[1];  // int64 edge_index [2,E]
    const float*     W    = (const float*)d_in[2];
    const float*     bias = (const float*)d_in[3];
    float*           out  = (float*)d_out;

    int n = in_sizes[0] / IN_CH;
    long long e = (long long)(in_sizes[1] / 2);
    const long long* row = ei;
    const long long* col = ei + e;

    // workspace: dis[N] | h1[N*128] | h2[N*128]  (~103 MB)
    float* ws = (float*)d_ws;
    size_t npad = ((size_t)n + 63) & ~(size_t)63;
    float* dis = ws;
    float* h1  = ws + npad;
    float* h2  = h1 + (size_t)n * IN_CH;

    // 1) symmetric GCN norm: deg (incl. self loop) -> deg^{-1/2}
    deg_init_kernel<<<(n + 255) / 256, 256, 0, stream>>>(dis, n);
    deg_accum_kernel<<<(int)((e + 255) / 256), 256, 0, stream>>>(col, dis, e);
    rsqrt_kernel<<<(n + 255) / 256, 256, 0, stream>>>(dis, n);

    // 2) two propagation rounds: h_{t+1}[c] = dis[c]^2 h_t[c] + sum_e dis[r]dis[c] h_t[r]
    long long tot4 = (long long)n * (IN_CH / 4);
    self_loop_kernel<<<(int)((tot4 + 255) / 256), 256, 0, stream>>>(x, dis, h1, n);
    spmm_edge_kernel<<<(int)((e + 7) / 8), 256, 0, stream>>>(row, col, dis, x, h1, e);
    self_loop_kernel<<<(int)((tot4 + 255) / 256), 256, 0, stream>>>(h1, dis, h2, n);
    spmm_edge_kernel<<<(int)((e + 7) / 8), 256, 0, stream>>>(row, col, dis, h1, h2, e);

    // 3) fused f32-WMMA GEMM + bias + log_softmax
    int tiles = (n + 15) / 16;
    gemm_lsm_kernel<<<(tiles + 3) / 4, 128, 0, stream>>>(h2, W, bias, out, n);
}